// EquivariantInterface_71193377898822
// MI455X (gfx1250) — compile-verified
//
#include <hip/hip_runtime.h>
#include <hip/hip_bf16.h>

#define DEVFN __device__ __forceinline__

typedef __attribute__((ext_vector_type(2))) float v2f;
typedef __attribute__((ext_vector_type(8))) float v8f;

#define BB 8192
#define EPSf 1e-6f
#define BN_EPSf 1e-5f
#define TAUf 0.01f

// ---------------- workspace layout (floats) ----------------
#define OFF_X0    2048                       // (B,2,3,5)
#define OFF_X1    (OFF_X0 + BB*30)           // (B,32,3,5)
#define OFF_X2    (OFF_X1 + BB*480)          // (B,32,3,5)
#define OFF_HCAT  (OFF_X2 + BB*480)          // (B,64,3,5)
#define OFF_Z4    (OFF_HCAT + BB*960)        // (B,32,3,5)
#define OFF_Z5    (OFF_Z4 + BB*480)          // (B,16,3,5)
#define OFF_Z0    (OFF_Z5 + BB*240)          // (B,3,3,5)
#define OFF_H2    (OFF_Z0 + BB*45)           // (B,32,80) stage-2 lrelu out (padded j)
#define OFF_PH    (OFF_H2 + BB*2560)         // (B,5)
#define OFF_PK    (OFF_PH + BB*5)            // (B,9)
#define OUT_KS    204800
#define OUT_ENT   278528

// ---------------- small helpers ----------------
DEVFN float sel5(float a0,float a1,float a2,float a3,float a4,int m){
  float r=a0; r=(m==1)?a1:r; r=(m==2)?a2:r; r=(m==3)?a3:r; r=(m==4)?a4:r; return r;
}

// dist row + k-th largest neighbor (stable ties -> lowest index, matches top_k)
DEVFN int knn_select(const float* xs, int CF, int nn, int kq){
  float sq[5]={0,0,0,0,0}, inn[5]={0,0,0,0,0};
  for (int cf=0; cf<CF; ++cf){
    float vn = xs[cf*5+nn];
    #pragma unroll
    for (int m=0;m<5;++m){ float v=xs[cf*5+m]; sq[m]+=v*v; inn[m]+=vn*v; }
  }
  float sqn = sel5(sq[0],sq[1],sq[2],sq[3],sq[4],nn);
  float dist[5];
  #pragma unroll
  for (int m=0;m<5;++m) dist[m] = 2.f*inn[m]-sqn-sq[m];
  int used=0,res=0;
  #pragma unroll
  for (int t=0;t<4;++t){
    float best=-3.4e38f; int bi=0;
    #pragma unroll
    for (int m=0;m<5;++m) if(!((used>>m)&1) && dist[m]>best){best=dist[m];bi=m;}
    if (t==kq) res=bi;
    used |= 1<<bi;
  }
  return res;
}

DEVFN void knn_row(const float* xs, int CF, int nn, int out[4]){
  float sq[5]={0,0,0,0,0}, inn[5]={0,0,0,0,0};
  for (int cf=0; cf<CF; ++cf){
    float vn = xs[cf*5+nn];
    #pragma unroll
    for (int m=0;m<5;++m){ float v=xs[cf*5+m]; sq[m]+=v*v; inn[m]+=vn*v; }
  }
  float sqn = sel5(sq[0],sq[1],sq[2],sq[3],sq[4],nn);
  float dist[5];
  #pragma unroll
  for (int m=0;m<5;++m) dist[m] = 2.f*inn[m]-sqn-sq[m];
  int used=0;
  #pragma unroll
  for (int t=0;t<4;++t){
    float best=-3.4e38f; int bi=0;
    #pragma unroll
    for (int m=0;m<5;++m) if(!((used>>m)&1) && dist[m]>best){best=dist[m];bi=m;}
    out[t]=bi; used |= 1<<bi;
  }
}

// --------- branchless F-column loaders (per-lane column fixed; clamped addr + mask) ---------
struct LoadContig {            // contiguous (CH,3,5) channel-major buffer
  const float* base; int j, nn; float msk;
  DEVFN float operator()(int k) const { return base[(k*3+j)*5+nn] * msk; }
};
struct LoadX12 {               // concat[x1;x2] along channels
  const float* x1s; const float* x2s; int j, nn; float msk;
  DEVFN float operator()(int k) const {
    const float* p = (k<32) ? x1s : x2s;
    int kk = (k<32) ? k : k-32;
    return p[(kk*3+j)*5+nn] * msk;
  }
};
struct LoadGF2 {               // graph-feature of x1: [nbr-ctr ; ctr]
  const float* x1s; int j, nn, mm; float msk;
  DEVFN float operator()(int k) const {
    if (k<32) return (x1s[(k*3+j)*5+mm] - x1s[(k*3+j)*5+nn]) * msk;
    return x1s[((k-32)*3+j)*5+nn] * msk;
  }
};

// ---------------- core WMMA GEMM: P(=Wf*F) [and D(=Wd*F)] -> LDS ----------------
// B-fragment (and A when single-weight) preloaded into registers so the
// v_wmma_f32_16x16x4_f32 sequence issues back-to-back (ISA §7.5).
template<int CIN, int MT, bool DUAL, class LD>
DEVFN void wave_wmma(const float* __restrict__ Wf, const float* __restrict__ Wd,
                     const LD& ld, float* ldsP, float* ldsD, int lane){
  const int half = lane >> 4;
  const int col  = lane & 15;
  float bx[CIN/2];
  #pragma unroll
  for (int s=0; s<CIN/4; ++s){
    const int kA = 4*s + half*2;
    bx[2*s]   = ld(kA);
    bx[2*s+1] = ld(kA+1);
  }
  v8f accP[MT]; v8f accD[MT];
  #pragma unroll
  for (int mt=0; mt<MT; ++mt){
    #pragma unroll
    for (int r=0;r<8;++r){ accP[mt][r]=0.f; if (DUAL) accD[mt][r]=0.f; }
  }
  if (!DUAL){
    float ax[MT][CIN/2];
    #pragma unroll
    for (int mt=0; mt<MT; ++mt){
      const int row = mt*16 + col;
      #pragma unroll
      for (int s=0; s<CIN/4; ++s){
        const int kA = 4*s + half*2;
        ax[mt][2*s]   = Wf[row*CIN + kA];
        ax[mt][2*s+1] = Wf[row*CIN + kA + 1];
      }
    }
    #pragma unroll
    for (int s=0; s<CIN/4; ++s){
      v2f bf; bf.x = bx[2*s]; bf.y = bx[2*s+1];
      #pragma unroll
      for (int mt=0; mt<MT; ++mt){
        v2f af; af.x = ax[mt][2*s]; af.y = ax[mt][2*s+1];
        accP[mt] = __builtin_amdgcn_wmma_f32_16x16x4_f32(false, af, false, bf,
                                                         (short)0, accP[mt], false, false);
      }
    }
  } else {
    #pragma unroll
    for (int s=0; s<CIN/4; ++s){
      const int kA = 4*s + half*2;
      v2f bf; bf.x = bx[2*s]; bf.y = bx[2*s+1];
      #pragma unroll
      for (int mt=0; mt<MT; ++mt){
        const int row = mt*16 + col;
        v2f af; af.x = Wf[row*CIN + kA]; af.y = Wf[row*CIN + kA + 1];
        accP[mt] = __builtin_amdgcn_wmma_f32_16x16x4_f32(false, af, false, bf,
                                                         (short)0, accP[mt], false, false);
        v2f ad; ad.x = Wd[row*CIN + kA]; ad.y = Wd[row*CIN + kA + 1];
        accD[mt] = __builtin_amdgcn_wmma_f32_16x16x4_f32(false, ad, false, bf,
                                                         (short)0, accD[mt], false, false);
      }
    }
  }
  #pragma unroll
  for (int mt=0; mt<MT; ++mt){
    #pragma unroll
    for (int r=0;r<8;++r){
      const int rowD = mt*16 + half*8 + r;   // C/D layout: M = r + 8*half (+16*mt)
      ldsP[rowD*16 + col] = accP[mt][r];
      if (DUAL) ldsD[rowD*16 + col] = accD[mt][r];
    }
  }
  (void)ldsD;
}

// BN stats accumulation over 5 groups of 3 (cols = nn*3+j)
DEVFN void stats_acc15(const float* P, int o, float* st){
  float s=0.f, s2=0.f;
  #pragma unroll
  for (int nn=0;nn<5;++nn){
    float q=0.f;
    #pragma unroll
    for (int j=0;j<3;++j){ float v=P[o*16+nn*3+j]; q+=v*v; }
    float nrm = sqrtf(q)+EPSf; s+=nrm; s2+=nrm*nrm;
  }
  atomicAdd(&st[o], s); atomicAdd(&st[64+o], s2);
}

// BN + VN-LeakyReLU epilogue, cols = nn*3+j, per-channel d
DEVFN void epilogue_std(const float* P, const float* D, const float* st,
                        const float* ga, const float* be, int o, float* outbase){
  float mean=st[128+o], inv=st[192+o], g=ga[o], b=be[o];
  #pragma unroll
  for (int nn=0;nn<5;++nn){
    float q=0.f;
    #pragma unroll
    for (int j=0;j<3;++j){ float v=P[o*16+nn*3+j]; q+=v*v; }
    float nrm = sqrtf(q)+EPSf;
    float scale = (g*(nrm-mean)*inv + b)/nrm;
    float dot=0.f,dsq=0.f, pb[3], dd[3];
    #pragma unroll
    for (int j=0;j<3;++j){
      float pv=P[o*16+nn*3+j]*scale; float dv=D[o*16+nn*3+j];
      dot+=pv*dv; dsq+=dv*dv; pb[j]=pv; dd[j]=dv;
    }
    float coef = (dot>=0.f)?0.f:(0.8f*dot/(dsq+EPSf));
    #pragma unroll
    for (int j=0;j<3;++j) outbase[j*5+nn] = pb[j] - coef*dd[j];
  }
}

// ================= kernels =================
__global__ void __launch_bounds__(256) k_init(float* ws, float* out){
  int t = blockIdx.x*blockDim.x + threadIdx.x;
  if (t < 1280) ws[t] = 0.f;
  if (t == 0) out[OUT_ENT] = 0.f;
}

__global__ void __launch_bounds__(256) k_x0(const float* __restrict__ nf,
                                            const float* __restrict__ noise,
                                            float* __restrict__ x0){
  int t = blockIdx.x*blockDim.x + threadIdx.x;
  if (t >= BB*30) return;
  int bb = t/30, r = t%30;
  int c = r/15, j = (r/5)%3, nn = r%5;
  int b = bb >> 6;
  float v = nf[((b*5+nn)*3+j)*2+c] + noise[((bb*5+nn)*3+j)*2+c];
  x0[bb*30 + (c*3+j)*5 + nn] = v;
}

__global__ void __launch_bounds__(64) k_bn_fin(float* st, float count){
  int c = threadIdx.x;
  if (c < 64){
    float mean = st[c]/count;
    float var  = st[64+c]/count - mean*mean;
    st[128+c] = mean;
    st[192+c] = 1.f/sqrtf(var + BN_EPSf);
  }
}

// ---- stage 1 (C=4, tiny: VALU, lane = channel) ----
__global__ void __launch_bounds__(128) k_s1_stats(const float* __restrict__ x0b,
                                                  const float* __restrict__ Wf1, float* st){
  int lane = threadIdx.x&31, wv = threadIdx.x>>5;
  int bb = blockIdx.x*4 + wv;
  const float* xs = x0b + bb*30;
  int idxA[5][4];
  #pragma unroll
  for (int nn=0;nn<5;++nn) knn_row(xs, 6, nn, idxA[nn]);
  int o = lane;
  float wf[4];
  #pragma unroll
  for (int c=0;c<4;++c) wf[c]=Wf1[o*4+c];
  float s=0.f,s2=0.f;
  #pragma unroll
  for (int nn=0;nn<5;++nn){
    #pragma unroll
    for (int kk=0;kk<4;++kk){
      int mm = idxA[nn][kk];
      float q=0.f;
      #pragma unroll
      for (int j=0;j<3;++j){
        float xa = xs[j*5+nn], xb = xs[(3+j)*5+nn];
        float f0 = xs[j*5+mm]-xa, f1 = xs[(3+j)*5+mm]-xb;
        float p = wf[0]*f0+wf[1]*f1+wf[2]*xa+wf[3]*xb;
        q += p*p;
      }
      float nrm = sqrtf(q)+EPSf; s+=nrm; s2+=nrm*nrm;
    }
  }
  atomicAdd(&st[o], s); atomicAdd(&st[64+o], s2);
}

__global__ void __launch_bounds__(128) k_s1_apply(const float* __restrict__ x0b,
    const float* __restrict__ Wf1, const float* __restrict__ Wd1,
    const float* __restrict__ gam, const float* __restrict__ bet,
    const float* __restrict__ Wp1, const float* __restrict__ st, float* __restrict__ x1out){
  __shared__ float H[4][1920];     // [wave][32ch][60]
  int lane = threadIdx.x&31, wv = threadIdx.x>>5;
  int bb = blockIdx.x*4 + wv;
  const float* xs = x0b + bb*30;
  int idxA[5][4];
  #pragma unroll
  for (int nn=0;nn<5;++nn) knn_row(xs, 6, nn, idxA[nn]);
  int o = lane;
  float wf[4], wd[4];
  #pragma unroll
  for (int c=0;c<4;++c){ wf[c]=Wf1[o*4+c]; wd[c]=Wd1[o*4+c]; }
  float mean=st[128+o], inv=st[192+o], g=gam[o], b=bet[o];
  #pragma unroll
  for (int nn=0;nn<5;++nn){
    #pragma unroll
    for (int kk=0;kk<4;++kk){
      int mm = idxA[nn][kk];
      float pv[3], dv[3], q=0.f;
      #pragma unroll
      for (int j=0;j<3;++j){
        float xa = xs[j*5+nn], xb = xs[(3+j)*5+nn];
        float f0 = xs[j*5+mm]-xa, f1 = xs[(3+j)*5+mm]-xb;
        pv[j] = wf[0]*f0+wf[1]*f1+wf[2]*xa+wf[3]*xb;
        dv[j] = wd[0]*f0+wd[1]*f1+wd[2]*xa+wd[3]*xb;
        q += pv[j]*pv[j];
      }
      float nrm=sqrtf(q)+EPSf;
      float scale=(g*(nrm-mean)*inv+b)/nrm;
      float dot=0.f,dsq=0.f;
      #pragma unroll
      for (int j=0;j<3;++j){ pv[j]*=scale; dot+=pv[j]*dv[j]; dsq+=dv[j]*dv[j]; }
      float coef=(dot>=0.f)?0.f:(0.8f*dot/(dsq+EPSf));
      #pragma unroll
      for (int j=0;j<3;++j) H[wv][o*60+(j*5+nn)*4+kk] = pv[j]-coef*dv[j];
    }
  }
  __syncthreads();
  const float* wp = Wp1 + o*32;
  #pragma unroll
  for (int nn=0;nn<5;++nn){
    float bestd=-3.4e38f; int bkk=0;
    #pragma unroll
    for (int kk=0;kk<4;++kk){
      float dv=0.f;
      #pragma unroll
      for (int j=0;j<3;++j){
        float dp=0.f;
        for (int c=0;c<32;++c) dp += wp[c]*H[wv][c*60+(j*5+nn)*4+kk];
        dv += H[wv][o*60+(j*5+nn)*4+kk]*dp;
      }
      if (dv>bestd){bestd=dv;bkk=kk;}
    }
    #pragma unroll
    for (int j=0;j<3;++j) x1out[(bb*32+o)*15 + j*5+nn] = H[wv][o*60+(j*5+nn)*4+bkk];
  }
}

// ---- stage 2: WMMA on gathered graph features (C=64, padded 80 cols, 5 tiles) ----
__global__ void __launch_bounds__(128) k_s2_stats(const float* __restrict__ x1b,
                                                  const float* __restrict__ Wf2, float* st){
  __shared__ float LP[4][512];
  int lane = threadIdx.x&31, wv = threadIdx.x>>5;
  int w = blockIdx.x*4 + wv;
  int bb = w/5, tile = w%5;
  const float* x1s = x1b + bb*480;
  __builtin_prefetch(x1s, 0, 3);
  __builtin_prefetch(x1s + 240, 0, 3);
  int colg = tile*16 + (lane&15);
  int grp = colg>>2, j = colg&3;
  int nn = grp>>2, kq = grp&3;
  int mm = knn_select(x1s, 96, nn, kq);
  float msk = (j<3) ? 1.f : 0.f;
  int jc = (j<3) ? j : 2;
  LoadGF2 ld{x1s, jc, nn, mm, msk};
  wave_wmma<64,2,false>(Wf2, (const float*)nullptr, ld, LP[wv], (float*)nullptr, lane);
  __syncthreads();
  int o = lane;
  float s=0.f,s2=0.f;
  #pragma unroll
  for (int g=0; g<4; ++g){
    float q=0.f;
    #pragma unroll
    for (int jj=0;jj<3;++jj){ float v=LP[wv][o*16+g*4+jj]; q+=v*v; }
    float nrm = sqrtf(q)+EPSf; s+=nrm; s2+=nrm*nrm;
  }
  atomicAdd(&st[o], s); atomicAdd(&st[64+o], s2);
}

__global__ void __launch_bounds__(128) k_s2_apply(const float* __restrict__ x1b,
    const float* __restrict__ Wf2, const float* __restrict__ Wd2,
    const float* __restrict__ gam, const float* __restrict__ bet,
    const float* __restrict__ st, float* __restrict__ h2){
  __shared__ float LP[4][512];
  __shared__ float LD_[4][512];
  int lane = threadIdx.x&31, wv = threadIdx.x>>5;
  int w = blockIdx.x*4 + wv;
  int bb = w/5, tile = w%5;
  const float* x1s = x1b + bb*480;
  __builtin_prefetch(x1s, 0, 3);
  __builtin_prefetch(x1s + 240, 0, 3);
  int colg = tile*16 + (lane&15);
  int grp = colg>>2, j = colg&3;
  int nn = grp>>2, kq = grp&3;
  int mm = knn_select(x1s, 96, nn, kq);
  float msk = (j<3) ? 1.f : 0.f;
  int jc = (j<3) ? j : 2;
  LoadGF2 ld{x1s, jc, nn, mm, msk};
  wave_wmma<64,2,true>(Wf2, Wd2, ld, LP[wv], LD_[wv], lane);
  __syncthreads();
  int o = lane;
  float mean=st[128+o], inv=st[192+o], g=gam[o], b=bet[o];
  float* hrow = h2 + (bb*32+o)*80 + tile*16;
  #pragma unroll
  for (int gg=0; gg<4; ++gg){
    float q=0.f;
    #pragma unroll
    for (int jj=0;jj<3;++jj){ float v=LP[wv][o*16+gg*4+jj]; q+=v*v; }
    float nrm = sqrtf(q)+EPSf;
    float scale = (g*(nrm-mean)*inv + b)/nrm;
    float dot=0.f,dsq=0.f, pb[3], dd[3];
    #pragma unroll
    for (int jj=0;jj<3;++jj){
      float pv=LP[wv][o*16+gg*4+jj]*scale; float dv=LD_[wv][o*16+gg*4+jj];
      dot+=pv*dv; dsq+=dv*dv; pb[jj]=pv; dd[jj]=dv;
    }
    float coef=(dot>=0.f)?0.f:(0.8f*dot/(dsq+EPSf));
    #pragma unroll
    for (int jj=0;jj<3;++jj) hrow[gg*4+jj] = pb[jj]-coef*dd[jj];
  }
}

__global__ void __launch_bounds__(128) k_pool2(const float* __restrict__ h2,
                                               const float* __restrict__ Wp2,
                                               float* __restrict__ x2out){
  __shared__ float H[4][1920];     // [wave][32ch][20grp*3j]
  int lane = threadIdx.x&31, wv = threadIdx.x>>5;
  int bb = blockIdx.x*4 + wv;
  int o = lane;
  const float* hr = h2 + (bb*32+o)*80;
  __builtin_prefetch(hr, 0, 3);
  #pragma unroll
  for (int grp=0; grp<20; ++grp)
    #pragma unroll
    for (int j=0;j<3;++j) H[wv][o*60 + grp*3+j] = hr[grp*4+j];
  __syncthreads();
  const float* wp = Wp2 + o*32;
  #pragma unroll
  for (int nn=0;nn<5;++nn){
    float bestd=-3.4e38f; int bkk=0;
    #pragma unroll
    for (int kk=0;kk<4;++kk){
      int grp = nn*4+kk;
      float dv=0.f;
      #pragma unroll
      for (int j=0;j<3;++j){
        float dp=0.f;
        for (int c=0;c<32;++c) dp += wp[c]*H[wv][c*60+grp*3+j];
        dv += H[wv][o*60+grp*3+j]*dp;
      }
      if (dv>bestd){bestd=dv;bkk=kk;}
    }
    #pragma unroll
    for (int j=0;j<3;++j) x2out[(bb*32+o)*15 + j*5+nn] = H[wv][o*60+(nn*4+bkk)*3+j];
  }
}

// ---- stage 3: Wf3 (32x64) on x12, Wd3 is 1x64, output hcat (h ; mean_n h) ----
__global__ void __launch_bounds__(128) k_s3_stats(const float* __restrict__ x1b,
    const float* __restrict__ x2b, const float* __restrict__ Wf3, float* st){
  __shared__ float LP[4][512];
  int lane = threadIdx.x&31, wv = threadIdx.x>>5;
  int bb = blockIdx.x*4 + wv;
  int col = lane&15; bool valid = col<15;
  int nn = valid? col/3 : 0, j = valid? col%3 : 0;
  LoadX12 ld{x1b+bb*480, x2b+bb*480, j, nn, valid?1.f:0.f};
  wave_wmma<64,2,false>(Wf3, (const float*)nullptr, ld, LP[wv], (float*)nullptr, lane);
  __syncthreads();
  stats_acc15(LP[wv], lane, st);
}

__global__ void __launch_bounds__(128) k_s3_apply(const float* __restrict__ x1b,
    const float* __restrict__ x2b, const float* __restrict__ Wf3,
    const float* __restrict__ Wd3, const float* __restrict__ gam,
    const float* __restrict__ bet, const float* __restrict__ st,
    float* __restrict__ hcat){
  __shared__ float LP[4][512];
  __shared__ float D3[4][16];
  int lane = threadIdx.x&31, wv = threadIdx.x>>5;
  int bb = blockIdx.x*4 + wv;
  int col = lane&15; bool valid = col<15;
  int nnc = valid? col/3 : 0, jc = valid? col%3 : 0;
  LoadX12 ld{x1b+bb*480, x2b+bb*480, jc, nnc, valid?1.f:0.f};
  wave_wmma<64,2,false>(Wf3, (const float*)nullptr, ld, LP[wv], (float*)nullptr, lane);
  if (lane < 15){
    int nn2 = lane/3, j2 = lane%3;
    const float* xa = x1b+bb*480; const float* xb = x2b+bb*480;
    float s=0.f;
    for (int c=0;c<32;++c) s += Wd3[c]*xa[(c*3+j2)*5+nn2];
    for (int c=0;c<32;++c) s += Wd3[32+c]*xb[(c*3+j2)*5+nn2];
    D3[wv][lane]=s;
  }
  __syncthreads();
  int o = lane;
  float mean=st[128+o], inv=st[192+o], g=gam[o], b=bet[o];
  float mj[3]={0.f,0.f,0.f};
  float* hb  = hcat + (bb*64+o)*15;
  float* hb2 = hcat + (bb*64+32+o)*15;
  #pragma unroll
  for (int nn=0;nn<5;++nn){
    float q=0.f;
    #pragma unroll
    for (int j=0;j<3;++j){ float v=LP[wv][o*16+nn*3+j]; q+=v*v; }
    float nrm = sqrtf(q)+EPSf;
    float scale = (g*(nrm-mean)*inv + b)/nrm;
    float dot=0.f,dsq=0.f, pb[3];
    #pragma unroll
    for (int j=0;j<3;++j){
      float pv=LP[wv][o*16+nn*3+j]*scale; float dv=D3[wv][nn*3+j];
      dot+=pv*dv; dsq+=dv*dv; pb[j]=pv;
    }
    float coef=(dot>=0.f)?0.f:(0.8f*dot/(dsq+EPSf));
    #pragma unroll
    for (int j=0;j<3;++j){ float h = pb[j]-coef*D3[wv][nn*3+j]; hb[j*5+nn]=h; mj[j]+=h; }
  }
  #pragma unroll
  for (int j=0;j<3;++j){
    float v = mj[j]*0.2f;
    #pragma unroll
    for (int nn=0;nn<5;++nn) hb2[j*5+nn]=v;
  }
}

// ---- generic contiguous-input WMMA stages (4 & 5) ----
template<int CIN, int MT, int OCH>
__global__ void __launch_bounds__(128) k_contig_stats(const float* __restrict__ inb,
    const float* __restrict__ Wf, float* st){
  __shared__ float LP[4][MT*256];
  int lane = threadIdx.x&31, wv = threadIdx.x>>5;
  int bb = blockIdx.x*4 + wv;
  int col = lane&15; bool valid = col<15;
  int nn = valid? col/3 : 0, j = valid? col%3 : 0;
  LoadContig ld{inb + bb*CIN*15, j, nn, valid?1.f:0.f};
  wave_wmma<CIN,MT,false>(Wf, (const float*)nullptr, ld, LP[wv], (float*)nullptr, lane);
  __syncthreads();
  if (lane < OCH) stats_acc15(LP[wv], lane, st);
}

template<int CIN, int MT, int OCH>
__global__ void __launch_bounds__(128) k_contig_apply(const float* __restrict__ inb,
    const float* __restrict__ Wf, const float* __restrict__ Wd,
    const float* __restrict__ gam, const float* __restrict__ bet,
    const float* __restrict__ st, float* __restrict__ outb){
  __shared__ float LP[4][MT*256];
  __shared__ float LD_[4][MT*256];
  int lane = threadIdx.x&31, wv = threadIdx.x>>5;
  int bb = blockIdx.x*4 + wv;
  int col = lane&15; bool valid = col<15;
  int nn = valid? col/3 : 0, j = valid? col%3 : 0;
  LoadContig ld{inb + bb*CIN*15, j, nn, valid?1.f:0.f};
  wave_wmma<CIN,MT,true>(Wf, Wd, ld, LP[wv], LD_[wv], lane);
  __syncthreads();
  if (lane < OCH)
    epilogue_std(LP[wv], LD_[wv], st, gam, bet, lane, outb + (bb*OCH+lane)*15);
}

// ---- Wslin: 3x16 ----
__global__ void __launch_bounds__(256) k_zlin(const float* __restrict__ z5,
    const float* __restrict__ Wslin, float* __restrict__ z0){
  int t = blockIdx.x*blockDim.x + threadIdx.x;
  if (t >= BB*45) return;
  int bb = t/45, r = t%45;
  int o = r/15, i = (r/5)%3, nn = r%5;
  float s=0.f;
  for (int c=0;c<16;++c) s += Wslin[o*16+c]*z5[(bb*16+c)*15 + i*5+nn];
  z0[bb*45 + (o*3+i)*5 + nn] = s;
}

// ---- score head: x_std/x12r/feat/pseudo_hs, wave per sample ----
__global__ void __launch_bounds__(128) k_head(const float* __restrict__ hcat,
    const float* __restrict__ x1b, const float* __restrict__ x2b,
    const float* __restrict__ z0, const float* __restrict__ Wh1,
    const float* __restrict__ bh1, float* __restrict__ ph){
  int lane = threadIdx.x&31, wv = threadIdx.x>>5;
  int bb = blockIdx.x*4 + wv;
  const float* z0s = z0 + bb*45;
  const float* hc  = hcat + bb*960;
  const float* x1s = x1b + bb*480;
  const float* x2s = x2b + bb*480;
  float accc = 0.f; float accm[5] = {0.f,0.f,0.f,0.f,0.f};
  #pragma unroll
  for (int t=0;t<2;++t){
    int i = lane + t*32;
    const float* hrow = hc + i*15;
    const float* xrow = (i<32) ? (x1s + i*15) : (x2s + (i-32)*15);
    #pragma unroll
    for (int k=0;k<3;++k){
      float w1 = Wh1[i*3+k], w2 = Wh1[192 + i*3+k];
      float rmax = -3.4e38f;
      #pragma unroll
      for (int m=0;m<5;++m){
        float xsv=0.f, xrv=0.f;
        #pragma unroll
        for (int j=0;j<3;++j){
          float z = z0s[(j*3+k)*5+m];
          xsv += hrow[j*5+m]*z; xrv += xrow[j*5+m]*z;
        }
        rmax = fmaxf(rmax, xsv);
        accm[m] += w2*xrv;
      }
      accc += w1*rmax;
    }
  }
  for (int off=16; off>0; off>>=1){
    accc += __shfl_down(accc, off, 32);
    #pragma unroll
    for (int m=0;m<5;++m) accm[m] += __shfl_down(accm[m], off, 32);
  }
  if (lane==0){
    float bv = bh1[0];
    #pragma unroll
    for (int m=0;m<5;++m) ph[bb*5+m] = accm[m] + accc + bv;
  }
}

// ---- Wp3 pool of x1 + Wh2 head -> pseudo_ks ----
__global__ void __launch_bounds__(128) k_pool3_ks(const float* __restrict__ x1b,
    const float* __restrict__ Wp3, const float* __restrict__ Wh2, float* __restrict__ pk){
  __shared__ float LX[4][480];
  __shared__ float LPOOL[4][96];
  int lane = threadIdx.x&31, wv = threadIdx.x>>5;
  int bb = blockIdx.x*4 + wv;
  int o = lane;
  const float* xr = x1b + (bb*32+o)*15;
  #pragma unroll
  for (int r=0;r<15;++r) LX[wv][o*15+r] = xr[r];
  __syncthreads();
  const float* wp = Wp3 + o*32;
  float bestd=-3.4e38f; int bnn=0;
  #pragma unroll
  for (int nn=0;nn<5;++nn){
    float dv=0.f;
    #pragma unroll
    for (int j=0;j<3;++j){
      float dval=0.f;
      for (int c=0;c<32;++c) dval += wp[c]*LX[wv][c*15+j*5+nn];
      dv += LX[wv][o*15+j*5+nn]*dval;
    }
    if (dv>bestd){bestd=dv;bnn=nn;}
  }
  #pragma unroll
  for (int j=0;j<3;++j) LPOOL[wv][o*3+j] = LX[wv][o*15+j*5+bnn];
  __syncthreads();
  if (lane < 9){
    int i = lane/3, oo = lane%3;
    float s=0.f;
    for (int c=0;c<32;++c) s += Wh2[oo*32+c]*LPOOL[wv][c*3+i];
    pk[bb*9 + i*3+oo] = s;
  }
}

// ---- Gram-Schmidt -> ks output ----
__global__ void __launch_bounds__(256) k_gs(const float* __restrict__ pk,
    const float* __restrict__ rot, float* __restrict__ out){
  int bb = blockIdx.x*blockDim.x + threadIdx.x;
  if (bb >= BB) return;
  float m[3][3];
  #pragma unroll
  for (int r=0;r<3;++r)
    #pragma unroll
    for (int c=0;c<3;++c) m[r][c] = pk[bb*9+r*3+c] + rot[bb*9+r*3+c];
  float e1[3], e2[3], e3[3];
  float n1 = sqrtf(m[0][0]*m[0][0]+m[1][0]*m[1][0]+m[2][0]*m[2][0]);
  #pragma unroll
  for (int r=0;r<3;++r) e1[r] = m[r][0]/(n1+1e-12f);
  float d12 = e1[0]*m[0][1]+e1[1]*m[1][1]+e1[2]*m[2][1];
  float v2[3];
  #pragma unroll
  for (int r=0;r<3;++r) v2[r] = m[r][1]-d12*e1[r];
  float n2 = sqrtf(v2[0]*v2[0]+v2[1]*v2[1]+v2[2]*v2[2]);
  #pragma unroll
  for (int r=0;r<3;++r) e2[r] = v2[r]/(n2+1e-12f);
  float d13 = e1[0]*m[0][2]+e1[1]*m[1][2]+e1[2]*m[2][2];
  float d23 = e2[0]*m[0][2]+e2[1]*m[1][2]+e2[2]*m[2][2];
  float v3[3];
  #pragma unroll
  for (int r=0;r<3;++r) v3[r] = m[r][2]-d13*e1[r]-d23*e2[r];
  float n3 = sqrtf(v3[0]*v3[0]+v3[1]*v3[1]+v3[2]*v3[2]);
  #pragma unroll
  for (int r=0;r<3;++r) e3[r] = v3[r]/(n3+1e-12f);
  #pragma unroll
  for (int r=0;r<3;++r){
    out[OUT_KS + bb*9 + r*3 + 0] = e1[r];
    out[OUT_KS + bb*9 + r*3 + 1] = e2[r];
    out[OUT_KS + bb*9 + r*3 + 2] = e3[r];
  }
}

// ---- scores + Sinkhorn + entropy + hard permutation ----
__global__ void __launch_bounds__(256) k_sinkhorn(const float* __restrict__ ph,
    const float* __restrict__ snoise, float* __restrict__ out){
  int bb = blockIdx.x*blockDim.x + threadIdx.x;
  if (bb >= BB) return;
  float s[5]; float q=0.f;
  #pragma unroll
  for (int m=0;m<5;++m){ s[m] = ph[bb*5+m] + snoise[bb*5+m]; q += s[m]*s[m]; }
  float nrm = fmaxf(sqrtf(q), 1e-12f);
  #pragma unroll
  for (int m=0;m<5;++m) s[m] /= nrm;
  int order[5]; float ss[5]; int used=0;
  #pragma unroll
  for (int t=0;t<5;++t){
    float best=-3.4e38f; int bi=0;
    #pragma unroll
    for (int m=0;m<5;++m) if(!((used>>m)&1) && s[m]>best){best=s[m];bi=m;}
    order[t]=bi; ss[t]=best; used |= 1<<bi;
  }
  float lp[5][5];
  #pragma unroll
  for (int i=0;i<5;++i)
    #pragma unroll
    for (int j=0;j<5;++j) lp[i][j] = -fabsf(s[i]-ss[j])*(1.f/TAUf);
  for (int it=0; it<20; ++it){
    #pragma unroll
    for (int i=0;i<5;++i){
      float mx=-3.4e38f;
      #pragma unroll
      for (int j=0;j<5;++j) mx = fmaxf(mx, lp[i][j]);
      float sm=0.f;
      #pragma unroll
      for (int j=0;j<5;++j) sm += expf(lp[i][j]-mx);
      float l = mx + logf(sm);
      #pragma unroll
      for (int j=0;j<5;++j) lp[i][j] -= l;
    }
    #pragma unroll
    for (int j=0;j<5;++j){
      float mx=-3.4e38f;
      #pragma unroll
      for (int i=0;i<5;++i) mx = fmaxf(mx, lp[i][j]);
      float sm=0.f;
      #pragma unroll
      for (int i=0;i<5;++i) sm += expf(lp[i][j]-mx);
      float l = mx + logf(sm);
      #pragma unroll
      for (int i=0;i<5;++i) lp[i][j] -= l;
    }
  }
  float pm[5][5];
  #pragma unroll
  for (int i=0;i<5;++i)
    #pragma unroll
    for (int j=0;j<5;++j) pm[i][j] = expf(lp[i][j]);
  float ent = 0.f;
  #pragma unroll
  for (int j=0;j<5;++j){
    float cs=0.f;
    #pragma unroll
    for (int i=0;i<5;++i) cs += fmaxf(pm[i][j],1e-12f);
    cs = fmaxf(cs,1e-12f);
    #pragma unroll
    for (int i=0;i<5;++i){
      float pr = fmaxf(pm[i][j],1e-12f)/cs;
      ent -= pr * fmaxf(logf(pr), -100.f);
    }
  }
  #pragma unroll
  for (int i=0;i<5;++i){
    float rs=0.f;
    #pragma unroll
    for (int j=0;j<5;++j) rs += fmaxf(pm[i][j],1e-12f);
    rs = fmaxf(rs,1e-12f);
    #pragma unroll
    for (int j=0;j<5;++j){
      float pr = fmaxf(pm[i][j],1e-12f)/rs;
      ent -= pr * fmaxf(logf(pr), -100.f);
    }
  }
  atomicAdd(out + OUT_ENT, ent * (1.f/40960.f));
  #pragma unroll
  for (int i=0;i<5;++i)
    #pragma unroll
    for (int j=0;j<5;++j) out[bb*25 + i*5 + j] = (order[j]==i) ? 1.f : 0.f;
}

// ================= host entry =================
extern "C" void kernel_launch(void* const* d_in, const int* in_sizes, int n_in,
                              void* d_out, int out_size, void* d_ws, size_t ws_size,
                              hipStream_t stream){
  (void)in_sizes; (void)n_in; (void)out_size; (void)ws_size;
  const float* nf    = (const float*)d_in[0];
  const float* noise = (const float*)d_in[1];
  const float* sn    = (const float*)d_in[2];
  const float* rot   = (const float*)d_in[3];
  const float *Wf1=(const float*)d_in[4], *Wd1=(const float*)d_in[5];
  const float *g1=(const float*)d_in[6],  *b1=(const float*)d_in[7];
  const float *Wf2=(const float*)d_in[8], *Wd2=(const float*)d_in[9];
  const float *g2=(const float*)d_in[10], *b2=(const float*)d_in[11];
  const float *Wf3=(const float*)d_in[12],*Wd3=(const float*)d_in[13];
  const float *g3=(const float*)d_in[14], *b3=(const float*)d_in[15];
  const float *Wp1=(const float*)d_in[16],*Wp2=(const float*)d_in[17],*Wp3=(const float*)d_in[18];
  const float *Ws1f=(const float*)d_in[19],*Ws1d=(const float*)d_in[20];
  const float *gs1=(const float*)d_in[21],*bs1=(const float*)d_in[22];
  const float *Ws2f=(const float*)d_in[23],*Ws2d=(const float*)d_in[24];
  const float *gs2=(const float*)d_in[25],*bs2=(const float*)d_in[26];
  const float *Wslin=(const float*)d_in[27],*Wh1=(const float*)d_in[28];
  const float *bh1=(const float*)d_in[29],*Wh2=(const float*)d_in[30];
  float* out = (float*)d_out;
  float* W   = (float*)d_ws;
  float* x0   = W + OFF_X0;
  float* x1   = W + OFF_X1;
  float* x2   = W + OFF_X2;
  float* hcat = W + OFF_HCAT;
  float* z4   = W + OFF_Z4;
  float* z5   = W + OFF_Z5;
  float* z0   = W + OFF_Z0;
  float* h2   = W + OFF_H2;
  float* ph   = W + OFF_PH;
  float* pk   = W + OFF_PK;

  k_init<<<5,256,0,stream>>>(W, out);
  k_x0<<<960,256,0,stream>>>(nf, noise, x0);

  // stage 1 (C=4)
  k_s1_stats<<<2048,128,0,stream>>>(x0, Wf1, W+0);
  k_bn_fin<<<1,64,0,stream>>>(W+0, (float)(BB*20));
  k_s1_apply<<<2048,128,0,stream>>>(x0, Wf1, Wd1, g1, b1, Wp1, W+0, x1);

  // stage 2 (C=64, WMMA on gathered graph features)
  k_s2_stats<<<10240,128,0,stream>>>(x1, Wf2, W+256);
  k_bn_fin<<<1,64,0,stream>>>(W+256, (float)(BB*20));
  k_s2_apply<<<10240,128,0,stream>>>(x1, Wf2, Wd2, g2, b2, W+256, h2);
  k_pool2<<<2048,128,0,stream>>>(h2, Wp2, x2);

  // stage 3 (Wf3 on x12, Wd3 1x64)
  k_s3_stats<<<2048,128,0,stream>>>(x1, x2, Wf3, W+512);
  k_bn_fin<<<1,64,0,stream>>>(W+512, (float)(BB*5));
  k_s3_apply<<<2048,128,0,stream>>>(x1, x2, Wf3, Wd3, g3, b3, W+512, hcat);

  // stage 4 (Ws1 64->32, WMMA)
  k_contig_stats<64,2,32><<<2048,128,0,stream>>>(hcat, Ws1f, W+768);
  k_bn_fin<<<1,64,0,stream>>>(W+768, (float)(BB*5));
  k_contig_apply<64,2,32><<<2048,128,0,stream>>>(hcat, Ws1f, Ws1d, gs1, bs1, W+768, z4);

  // stage 5 (Ws2 32->16, WMMA)
  k_contig_stats<32,1,16><<<2048,128,0,stream>>>(z4, Ws2f, W+1024);
  k_bn_fin<<<1,64,0,stream>>>(W+1024, (float)(BB*5));
  k_contig_apply<32,1,16><<<2048,128,0,stream>>>(z4, Ws2f, Ws2d, gs2, bs2, W+1024, z5);

  // heads
  k_zlin<<<1440,256,0,stream>>>(z5, Wslin, z0);
  k_head<<<2048,128,0,stream>>>(hcat, x1, x2, z0, Wh1, bh1, ph);
  k_pool3_ks<<<2048,128,0,stream>>>(x1, Wp3, Wh2, pk);
  k_gs<<<32,256,0,stream>>>(pk, rot, out);
  k_sinkhorn<<<32,256,0,stream>>>(ph, sn, out);
}